// StructureAwareImplicitGraphLearning_28132035789272
// MI455X (gfx1250) — compile-verified
//
#include <hip/hip_runtime.h>
#include <hip/hip_bf16.h>
#include <math.h>

// ---------------------------------------------------------------------------
// StructureAwareImplicitGraphLearning for MI455X (gfx1250, wave32, WMMA).
// All GEMMs use V_WMMA_F32_16X16X4_F32 (full fp32 precision).
// ---------------------------------------------------------------------------

#define NROW   1500
#define NPAD   1504            // 94 * 16
#define MT     (NPAD / 16)     // 94 row tiles
#define DDIM   64
#define TOPK   16
#define ALPHA  0.3f
#define LN_EPS 1e-5f

typedef __attribute__((ext_vector_type(2))) float v2f;
typedef __attribute__((ext_vector_type(8))) float v8f;

// ---------------------------------------------------------------------------
// Generic fp32 WMMA GEMM:  C[M x Ncols] = A[M x K] @ B (+bias, opt relu)
//   BTRANS == 0 : B stored row-major [K x Ncols], LDB = column stride
//   BTRANS == 1 : B stored as W[Ncols x K] (computes A @ W^T), LDB = K stride
// K and LDB are compile-time so the K-loop fully unrolls into a straight
// stream of global_load_b64 + v_wmma with immediate offsets (no address math
// in the hot path). Block = 128 threads (4 waves); wave w handles column tile
// blockIdx.y*4 + w. Ncols must be a multiple of 64; K a multiple of 4.
// Rows >= M: A addresses are clamped (row m of D depends only on row m of A,
// so garbage rows are harmless) and stores are guarded.
// ---------------------------------------------------------------------------
template <int BTRANS, int K, int LDB>
__global__ void wmma_gemm_f32_kernel(const float* __restrict__ A, int lda,
                                     const float* __restrict__ B,
                                     const float* __restrict__ bias,
                                     float* __restrict__ C, int ldc,
                                     int M, int relu_act)
{
    const int wave  = threadIdx.x >> 5;
    const int lane  = threadIdx.x & 31;
    const int half  = lane >> 4;          // 0: K pair {0,1}, 1: K pair {2,3}
    const int l16   = lane & 15;

    const int tileM = blockIdx.x;
    const int tileN = blockIdx.y * 4 + wave;

    const int rowA  = tileM * 16 + l16;              // A row sourced by this lane
    const int rowC  = rowA < M ? rowA : (M - 1);     // clamped (OOB-safe)
    const int colB  = tileN * 16 + l16;              // B column sourced by this lane

    const float* Ap = A + (size_t)rowC * lda + 2 * half;   // 8B-aligned
    const float* Bp;
    if (BTRANS) Bp = B + (size_t)colB * LDB + 2 * half;    // contiguous pair
    else        Bp = B + (size_t)(2 * half) * LDB + colB;  // strided pair

    v8f acc = {};
#pragma unroll
    for (int k = 0; k < K; k += 4) {
        v2f a = *(const v2f*)(Ap + k);
        v2f b;
        if (BTRANS) {
            b = *(const v2f*)(Bp + k);
        } else {
            b.x = Bp[k * LDB];          // compile-time offsets
            b.y = Bp[k * LDB + LDB];
        }
        // D = A(16x4) * B(4x16) + C   -- v_wmma_f32_16x16x4_f32
        acc = __builtin_amdgcn_wmma_f32_16x16x4_f32(
                  false, a, false, b, (short)0, acc, false, false);
    }

    const float bv = bias ? bias[tileN * 16 + l16] : 0.0f;
#pragma unroll
    for (int r = 0; r < 8; ++r) {
        const int row = tileM * 16 + r + 8 * half;   // C layout: vgpr r -> M = r (+8 hi lanes)
        if (row < M) {
            float v = acc[r] + bv;
            if (relu_act) v = fmaxf(v, 0.0f);
            C[(size_t)row * ldc + tileN * 16 + l16] = v;
        }
    }
}

// ---------------------------------------------------------------------------
// LayerNorm (over 64 dims) + ReLU, one wave32 per row, shfl reductions.
// ---------------------------------------------------------------------------
__global__ void ln_relu_kernel(const float* __restrict__ h0,
                               const float* __restrict__ g,
                               const float* __restrict__ b,
                               float* __restrict__ h, int M)
{
    const int row  = blockIdx.x * (blockDim.x >> 5) + (threadIdx.x >> 5);
    const int lane = threadIdx.x & 31;
    if (row >= M) return;

    const float x0 = h0[row * DDIM + lane];
    const float x1 = h0[row * DDIM + 32 + lane];

    float s = x0 + x1;
#pragma unroll
    for (int off = 16; off; off >>= 1) s += __shfl_xor(s, off, 32);
    const float mu = s * (1.0f / 64.0f);

    const float d0 = x0 - mu, d1 = x1 - mu;
    float v = d0 * d0 + d1 * d1;
#pragma unroll
    for (int off = 16; off; off >>= 1) v += __shfl_xor(v, off, 32);
    const float rsig = rsqrtf(v * (1.0f / 64.0f) + LN_EPS);

    h[row * DDIM + lane]      = fmaxf(d0 * rsig * g[lane]      + b[lane],      0.0f);
    h[row * DDIM + 32 + lane] = fmaxf(d1 * rsig * g[32 + lane] + b[32 + lane], 0.0f);
}

// ---------------------------------------------------------------------------
// Attention row + top-16 + masked/normalized pos/neg weights.
// One block (256 threads) per row i. Never materializes (N,N,64) `pre`.
//   att[i][j] = sigmoid( sum_k relu(pi[i][k] + pj[j][k]) * w2[k] + b2 )
// (att_b1 already folded into pi.) Diagonal excluded (sigmoid > 0 off-diag,
// so the zeroed diagonal is never in the reference top-k either).
// ---------------------------------------------------------------------------
__global__ void att_topk_kernel(const float* __restrict__ pi,
                                const float* __restrict__ pj,
                                const float* __restrict__ w2,
                                const float* __restrict__ b2,
                                const float* __restrict__ gflow,
                                int*   __restrict__ topk_idx,
                                float* __restrict__ wpos,
                                float* __restrict__ wneg)
{
    __shared__ float s_pi[DDIM];
    __shared__ float s_w2[DDIM];
    __shared__ float s_att[NROW];
    __shared__ float s_rv[256];
    __shared__ int   s_ri[256];
    __shared__ float s_sel_v[TOPK];
    __shared__ int   s_sel_i[TOPK];

    const int i = blockIdx.x;
    const int t = threadIdx.x;

    if (t < DDIM) {
        s_pi[t] = pi[i * DDIM + t];
        s_w2[t] = w2[t];
    }
    __syncthreads();

    const float b2v = b2[0];
    for (int j = t; j < NROW; j += 256) {
        const float* pjr = pj + j * DDIM;
        float acc = 0.0f;
#pragma unroll
        for (int k = 0; k < DDIM; ++k) {
            const float pre = fmaxf(s_pi[k] + pjr[k], 0.0f);
            acc = fmaf(pre, s_w2[k], acc);
        }
        const float a = 1.0f / (1.0f + __expf(-(acc + b2v)));
        s_att[j] = (j == i) ? -1.0f : a;
    }
    __syncthreads();

    // Iterative top-16 argmax over the row.
    for (int s = 0; s < TOPK; ++s) {
        float bv = -2.0f; int bi = 0;
        for (int j = t; j < NROW; j += 256) {
            const float v = s_att[j];
            if (v > bv) { bv = v; bi = j; }
        }
        s_rv[t] = bv; s_ri[t] = bi;
        __syncthreads();
#pragma unroll
        for (int off = 128; off; off >>= 1) {
            if (t < off) {
                if (s_rv[t + off] > s_rv[t]) { s_rv[t] = s_rv[t + off]; s_ri[t] = s_ri[t + off]; }
            }
            __syncthreads();
        }
        if (t == 0) {
            s_sel_v[s] = s_rv[0];
            s_sel_i[s] = s_ri[0];
            s_att[s_ri[0]] = -1.0f;   // remove from further rounds
        }
        __syncthreads();
    }

    if (t == 0) {
        float vp[TOPK], vn[TOPK];
        float sp = 0.0f, sn = 0.0f;
        for (int s = 0; s < TOPK; ++s) {
            const int   j = s_sel_i[s];
            const float a = s_sel_v[s];
            const float gv = gflow[(size_t)i * NROW + j];
            const float p = (gv >  ALPHA) ? a : 0.0f;
            const float q = (gv < -ALPHA) ? a : 0.0f;
            vp[s] = p; vn[s] = q; sp += p; sn += q;
        }
        const float rp = 1.0f / (sp + 1e-8f);
        const float rn = 1.0f / (sn + 1e-8f);
        for (int s = 0; s < TOPK; ++s) {
            topk_idx[i * TOPK + s] = s_sel_i[s];
            wpos[i * TOPK + s]     = vp[s] * rp;
            wneg[i * TOPK + s]     = vn[s] * rn;
        }
    }
}

// ---------------------------------------------------------------------------
// Sparse message gather:  m[i] = sum_s w[i][s] * feat[idx[i][s]]
// Block per row, 128 threads: lanes <64 do pos, >=64 do neg.
// ---------------------------------------------------------------------------
__global__ void gather_msg_kernel(const float* __restrict__ mp,
                                  const float* __restrict__ mn,
                                  const int*   __restrict__ idx,
                                  const float* __restrict__ wpos,
                                  const float* __restrict__ wneg,
                                  float* __restrict__ m_pos,
                                  float* __restrict__ m_neg)
{
    const int i   = blockIdx.x;
    const int t   = threadIdx.x;
    const int d   = t & 63;
    const bool nb = (t >= 64);
    const float* src = nb ? mn   : mp;
    const float* w   = nb ? wneg : wpos;
    float acc = 0.0f;
#pragma unroll
    for (int s = 0; s < TOPK; ++s) {
        const int j = idx[i * TOPK + s];
        acc = fmaf(w[i * TOPK + s], src[j * DDIM + d], acc);
    }
    (nb ? m_neg : m_pos)[i * DDIM + d] = acc;
}

// ---------------------------------------------------------------------------
// GRU gate fusion: h' = (1-z)*n + z*h, written into hcat[:, off : off+64].
// gi/gh already include their biases (added by the GEMM kernel).
// ---------------------------------------------------------------------------
__global__ void gru_kernel(const float* __restrict__ gi,
                           const float* __restrict__ gh,
                           const float* __restrict__ h,
                           float* __restrict__ hcat,
                           int M, int outOff)
{
    const int idx = blockIdx.x * blockDim.x + threadIdx.x;
    if (idx >= M * DDIM) return;
    const int row = idx >> 6;
    const int d   = idx & 63;

    const float ir = gi[row * 192 + d];
    const float iz = gi[row * 192 + 64 + d];
    const float in = gi[row * 192 + 128 + d];
    const float hr = gh[row * 192 + d];
    const float hz = gh[row * 192 + 64 + d];
    const float hn = gh[row * 192 + 128 + d];

    const float r = 1.0f / (1.0f + __expf(-(ir + hr)));
    const float z = 1.0f / (1.0f + __expf(-(iz + hz)));
    const float n = tanhf(in + r * hn);
    const float hv = h[row * DDIM + d];

    hcat[row * 128 + outOff + d] = (1.0f - z) * n + z * hv;
}

// ---------------------------------------------------------------------------
// Host-side launcher
// ---------------------------------------------------------------------------
extern "C" void kernel_launch(void* const* d_in, const int* in_sizes, int n_in,
                              void* d_out, int out_size, void* d_ws, size_t ws_size,
                              hipStream_t stream)
{
    const float* x_risk   = (const float*)d_in[0];   // (1,8,1500,64)
    const float* gflow    = (const float*)d_in[1];   // (1,1500,1500)
    const float* enc_w    = (const float*)d_in[2];   // (64,64)
    const float* enc_b    = (const float*)d_in[3];
    const float* ln_g     = (const float*)d_in[4];
    const float* ln_b     = (const float*)d_in[5];
    const float* att_w1   = (const float*)d_in[6];   // (128,64)
    const float* att_b1   = (const float*)d_in[7];   // (64,)
    const float* att_w2   = (const float*)d_in[8];   // (64,1)
    const float* att_b2   = (const float*)d_in[9];   // (1,)
    const float* msgp_w   = (const float*)d_in[10];  // (64,64)
    const float* msgp_b   = (const float*)d_in[11];
    const float* gp_wih   = (const float*)d_in[12];  // (192,64)
    const float* gp_whh   = (const float*)d_in[13];  // (192,64)
    const float* gp_bih   = (const float*)d_in[14];  // (192,)
    const float* gp_bhh   = (const float*)d_in[15];
    const float* msgn_w   = (const float*)d_in[16];
    const float* msgn_b   = (const float*)d_in[17];
    const float* gn_wih   = (const float*)d_in[18];
    const float* gn_whh   = (const float*)d_in[19];
    const float* gn_bih   = (const float*)d_in[20];
    const float* gn_bhh   = (const float*)d_in[21];
    const float* comb_w   = (const float*)d_in[22];  // (128,64)
    const float* comb_b   = (const float*)d_in[23];
    float* out = (float*)d_out;                      // (1,1500,64)

    // -------- workspace layout (fp32) --------
    float* W = (float*)d_ws;
    size_t off = 0;
    float* h0     = W + off; off += (size_t)NPAD * 64;
    float* h      = W + off; off += (size_t)NPAD * 64;
    float* pi     = W + off; off += (size_t)NPAD * 64;
    float* pj     = W + off; off += (size_t)NPAD * 64;
    float* mp     = W + off; off += (size_t)NPAD * 64;
    float* mn     = W + off; off += (size_t)NPAD * 64;
    float* m_pos  = W + off; off += (size_t)NPAD * 64;
    float* m_neg  = W + off; off += (size_t)NPAD * 64;
    float* gi_pos = W + off; off += (size_t)NPAD * 192;
    float* gh_pos = W + off; off += (size_t)NPAD * 192;
    float* gi_neg = W + off; off += (size_t)NPAD * 192;
    float* gh_neg = W + off; off += (size_t)NPAD * 192;
    float* hcat   = W + off; off += (size_t)NPAD * 128;
    int*   tkidx  = (int*)(W + off); off += (size_t)NROW * TOPK;
    float* wpos   = W + off; off += (size_t)NROW * TOPK;
    float* wneg   = W + off; off += (size_t)NROW * TOPK;

    const float* x_last = x_risk + (size_t)7 * NROW * DDIM;  // t = T-1

    const dim3 blkG(128);
    const dim3 grd64(MT, 1);    // Ncols = 64
    const dim3 grd192(MT, 3);   // Ncols = 192

    // 1) h0 = x_last @ enc_w + enc_b
    wmma_gemm_f32_kernel<0, 64, 64><<<grd64, blkG, 0, stream>>>(
        x_last, 64, enc_w, enc_b, h0, 64, NROW, 0);

    // 2) h = relu(LN(h0) * g + b)
    {
        const int waves_per_blk = 8;  // 256 threads
        const int nblk = (NROW + waves_per_blk - 1) / waves_per_blk;
        ln_relu_kernel<<<nblk, 256, 0, stream>>>(h0, ln_g, ln_b, h, NROW);
    }

    // 3) pi = h @ w1a + att_b1 ; pj = h @ w1b   (b1 folded into pi)
    wmma_gemm_f32_kernel<0, 64, 64><<<grd64, blkG, 0, stream>>>(
        h, 64, att_w1, att_b1, pi, 64, NROW, 0);
    wmma_gemm_f32_kernel<0, 64, 64><<<grd64, blkG, 0, stream>>>(
        h, 64, att_w1 + 64 * 64, nullptr, pj, 64, NROW, 0);

    // 4) per-node message features
    wmma_gemm_f32_kernel<0, 64, 64><<<grd64, blkG, 0, stream>>>(
        h, 64, msgp_w, msgp_b, mp, 64, NROW, 0);
    wmma_gemm_f32_kernel<0, 64, 64><<<grd64, blkG, 0, stream>>>(
        h, 64, msgn_w, msgn_b, mn, 64, NROW, 0);

    // 5) attention + top-16 + masked/normalized weights
    att_topk_kernel<<<NROW, 256, 0, stream>>>(
        pi, pj, att_w2, att_b2, gflow, tkidx, wpos, wneg);

    // 6) sparse message aggregation
    gather_msg_kernel<<<NROW, 128, 0, stream>>>(
        mp, mn, tkidx, wpos, wneg, m_pos, m_neg);

    // 7) GRU gate GEMMs (B transposed: x @ w.T + bias)
    wmma_gemm_f32_kernel<1, 64, 64><<<grd192, blkG, 0, stream>>>(
        m_pos, 64, gp_wih, gp_bih, gi_pos, 192, NROW, 0);
    wmma_gemm_f32_kernel<1, 64, 64><<<grd192, blkG, 0, stream>>>(
        h, 64, gp_whh, gp_bhh, gh_pos, 192, NROW, 0);
    wmma_gemm_f32_kernel<1, 64, 64><<<grd192, blkG, 0, stream>>>(
        m_neg, 64, gn_wih, gn_bih, gi_neg, 192, NROW, 0);
    wmma_gemm_f32_kernel<1, 64, 64><<<grd192, blkG, 0, stream>>>(
        h, 64, gn_whh, gn_bhh, gh_neg, 192, NROW, 0);

    // 8) GRU fusion -> hcat = [h_pos | h_neg]
    {
        const int total = NROW * DDIM;
        const int nblk = (total + 255) / 256;
        gru_kernel<<<nblk, 256, 0, stream>>>(gi_pos, gh_pos, h, hcat, NROW, 0);
        gru_kernel<<<nblk, 256, 0, stream>>>(gi_neg, gh_neg, h, hcat, NROW, 64);
    }

    // 9) out = hcat @ comb_w + comb_b
    wmma_gemm_f32_kernel<0, 128, 64><<<grd64, blkG, 0, stream>>>(
        hcat, 128, comb_w, comb_b, out, 64, NROW, 0);

    (void)in_sizes; (void)n_in; (void)out_size; (void)ws_size;
}